// PNN_52802327937174
// MI455X (gfx1250) — compile-verified
//
#include <hip/hip_runtime.h>
#include <hip/hip_bf16.h>
#include <math.h>

typedef __attribute__((ext_vector_type(2))) float v2f;
typedef __attribute__((ext_vector_type(8))) float v8f;

#define BATCH    8192
#define VOCAB    100000
#define EDIM     64
#define ND       15
#define SEQLEN   50
#define NF       16
#define NPAIR    120
#define HID      30
#define LZDIM    (NF * EDIM)           // 1024
#define INDIM    (LZDIM + NPAIR)       // 1144
#define XSTR     1156                  // s_x row stride: 1156 % 64 == 4 -> conflict-free
#define W1TSTR   1152                  // padded K stride for transposed W1 (8B aligned)
#define NCOLP    32                    // HID padded to 2 WMMA tiles
#define TILE_B   16
#define NTHREADS 256

// ---------------------------------------------------------------------------
// Prep kernel: W1 (1144 x 30, row-major) -> W1t (32 x 1152) transposed, padded
// with zeros. Makes the WMMA B-fragment an unconditional contiguous b64 load.
// ---------------------------------------------------------------------------
__global__ __launch_bounds__(256) void pad_w1_kernel(
    const float* __restrict__ W1, float* __restrict__ W1t)
{
    const int idx = blockIdx.x * blockDim.x + threadIdx.x;
    if (idx >= NCOLP * W1TSTR) return;
    const int col = idx / W1TSTR;
    const int k   = idx % W1TSTR;
    W1t[idx] = (col < HID && k < INDIM) ? W1[(size_t)k * HID + col] : 0.0f;
}

// ---------------------------------------------------------------------------
// Fused PNN kernel: one block = 16 samples.
// ---------------------------------------------------------------------------
__global__ __launch_bounds__(NTHREADS) void pnn_fused_kernel(
    const int* __restrict__ dense_ids,     // [B, ND]
    const int* __restrict__ seq_ids,       // [B, SEQLEN]
    const float* __restrict__ tables_dense,// [ND, V, D]
    const float* __restrict__ table_seq,   // [V, D]
    const float* __restrict__ inner_w,     // [NPAIR, D]
    const float* __restrict__ W1t,         // [32, 1152] transposed+padded
    const float* __restrict__ b1,          // [HID]
    const float* __restrict__ W2,          // [HID, 1]
    const float* __restrict__ b2,          // [1]
    float* __restrict__ out)               // [B, 1]
{
    __shared__ float s_x[TILE_B][XSTR];       // 72 KB: x[s][0..1143] contiguous
    __shared__ float s_iw[NPAIR][EDIM];       // 30 KB
    __shared__ float s_h[TILE_B][NCOLP];      //  2 KB

    const int tid = threadIdx.x;
    const int s0  = blockIdx.x * TILE_B;

    // -------- Phase 0: stage inner_w into LDS (B128 loads) -------------------
    {
        const float4* src = (const float4*)inner_w;
        float4* dst = (float4*)&s_iw[0][0];
        for (int i = tid; i < (NPAIR * EDIM) / 4; i += NTHREADS)
            dst[i] = src[i];
    }

    // -------- Phase 1a: dense embedding gathers (B128 per lane) --------------
    // 240 rows x 64 floats; 16 threads/row, each moves one float4.
    for (int i = tid; i < (TILE_B * ND) * 16; i += NTHREADS) {
        const int row   = i >> 4;
        const int lane4 = i & 15;
        const int s = row / ND;
        const int f = row % ND;
        const int id = dense_ids[(s0 + s) * ND + f];
        const float4 v = ((const float4*)(tables_dense
                          + (size_t)f * VOCAB * EDIM + (size_t)id * EDIM))[lane4];
        *(float4*)&s_x[s][f * EDIM + lane4 * 4] = v;
    }

    // -------- Phase 1b: sequence mean (table L2-resident: 25.6 MB) -----------
    {
        const int s  = tid >> 4;
        const int d4 = tid & 15;
        float4 acc = make_float4(0.f, 0.f, 0.f, 0.f);
        const int* sid = seq_ids + (s0 + s) * SEQLEN;
        for (int j = 0; j < SEQLEN; ++j) {
            const int id = sid[j];
            if (j + 1 < SEQLEN)
                __builtin_prefetch(table_seq + (size_t)sid[j + 1] * EDIM, 0, 1);
            const float4 v = ((const float4*)(table_seq + (size_t)id * EDIM))[d4];
            acc.x += v.x; acc.y += v.y; acc.z += v.z; acc.w += v.w;
        }
        const float inv = 1.0f / (float)SEQLEN;
        *(float4*)&s_x[s][ND * EDIM + d4 * 4] =
            make_float4(acc.x * inv, acc.y * inv, acc.z * inv, acc.w * inv);
    }
    __syncthreads();

    // -------- Phase 2: weighted pairwise inner products ----------------------
    for (int i = tid; i < NPAIR * TILE_B; i += NTHREADS) {
        const int p = i >> 4;
        const int s = i & 15;
        int r = 0, rem = p, span = NF - 1;
        while (rem >= span) { rem -= span; ++r; --span; }
        const int c = r + 1 + rem;
        const float* xr = &s_x[s][r * EDIM];
        const float* xc = &s_x[s][c * EDIM];
        float dot = 0.f;
        #pragma unroll 8
        for (int d = 0; d < EDIM; ++d)
            dot += xr[d] * xc[d] * s_iw[p][d];
        s_x[s][LZDIM + p] = dot;
    }
    __syncthreads();

    // -------- Phase 3: x(16x1144) @ W1(1144x32) via V_WMMA_F32_16X16X4_F32 ---
    // Waves 0,1 each own one 16-column tile; guard is wave-uniform -> EXEC all-1s.
    if (tid < 64) {
        const int lane  = tid & 31;
        const int ntile = tid >> 5;
        const int M     = lane & 15;                 // A-frag row
        const int kb    = (lane >> 4) * 2;           // A/B frag K sub-offset
        const int col   = ntile * 16 + (lane & 15);  // B-frag column (0..31)

        const float* xrow = &s_x[M][0];              // 8B-aligned at even k
        const float* wrow = W1t + (size_t)col * W1TSTR;

        // software-pipelined: fetch fragments one step ahead of the WMMA
        float2 af = *(const float2*)(xrow + kb);
        float2 bf = *(const float2*)(wrow + kb);
        v8f acc = {0.f, 0.f, 0.f, 0.f, 0.f, 0.f, 0.f, 0.f};
        for (int k0 = 4; k0 < INDIM; k0 += 4) {
            const float2 an = *(const float2*)(xrow + k0 + kb);
            const float2 bn = *(const float2*)(wrow + k0 + kb);
            v2f a; a.x = af.x; a.y = af.y;
            v2f b; b.x = bf.x; b.y = bf.y;
            acc = __builtin_amdgcn_wmma_f32_16x16x4_f32(
                false, a, false, b, (short)0, acc, false, false);
            af = an; bf = bn;
        }
        {
            v2f a; a.x = af.x; a.y = af.y;
            v2f b; b.x = bf.x; b.y = bf.y;
            acc = __builtin_amdgcn_wmma_f32_16x16x4_f32(
                false, a, false, b, (short)0, acc, false, false);
        }

        // C/D layout: VGPR r -> row M = r + 8*(lane>>4), column = col
        const bool colv = (col < HID);
        const float bias = colv ? b1[col] : 0.f;
        const int mhi = (lane >> 4) * 8;
        #pragma unroll
        for (int r = 0; r < 8; ++r) {
            float h = acc[r] + bias;
            h = (h > 0.f && colv) ? h : 0.f;         // ReLU + column mask
            s_h[mhi + r][col] = h;
        }
    }
    __syncthreads();

    // -------- Phase 4: (30 -> 1) + sigmoid -----------------------------------
    if (tid < TILE_B) {
        float acc = b2[0];
        #pragma unroll
        for (int j = 0; j < HID; ++j)
            acc += s_h[tid][j] * W2[j];
        out[s0 + tid] = 1.0f / (1.0f + __expf(-acc));
    }
}

extern "C" void kernel_launch(void* const* d_in, const int* in_sizes, int n_in,
                              void* d_out, int out_size, void* d_ws, size_t ws_size,
                              hipStream_t stream) {
    (void)in_sizes; (void)n_in; (void)out_size; (void)ws_size;
    const int*   dense_ids    = (const int*)  d_in[0];
    const int*   seq_ids      = (const int*)  d_in[1];
    const float* tables_dense = (const float*)d_in[2];
    const float* table_seq    = (const float*)d_in[3];
    const float* inner_w      = (const float*)d_in[4];
    const float* W1           = (const float*)d_in[5];
    const float* b1           = (const float*)d_in[6];
    const float* W2           = (const float*)d_in[7];
    const float* b2           = (const float*)d_in[8];
    float* out = (float*)d_out;
    float* W1t = (float*)d_ws;   // 32 * 1152 * 4 = 147456 bytes of scratch

    // Prep: transpose + zero-pad W1 into scratch (L2-resident thereafter).
    {
        const int n = NCOLP * W1TSTR;
        pad_w1_kernel<<<(n + 255) / 256, 256, 0, stream>>>(W1, W1t);
    }
    pnn_fused_kernel<<<dim3(BATCH / TILE_B), dim3(NTHREADS), 0, stream>>>(
        dense_ids, seq_ids, tables_dense, table_seq, inner_w,
        W1t, b1, W2, b2, out);
}